// PtrNet_80530636800324
// MI455X (gfx1250) — compile-verified
//
#include <hip/hip_runtime.h>
#include <math.h>

// ---------------- problem constants ----------------
constexpr int HID  = 256;    // l
constexpr int SEQ  = 1024;   // p
constexpr int SEQ1 = 1025;   // p+1
constexpr int X2L  = 512;    // 2l
constexpr int NBLK = 32;     // cooperative blocks (one per WGP)
constexpr int T    = 256;    // threads per block (8 waves)

// LDS pitches (floats). 520/264: odd-multiple-of-8 banks + 16B-aligned rows
constexpr int W2P = 264, WIP = 520, WHP = 264, VHP = 264, HP = 33;
constexpr int SMEM_FLOATS =
    8 * W2P +      // sW2T  : 8 COLUMNS of W2 (transposed slice)
    32 * WIP +     // sWih  : 32 gate rows (i,f,g,o for 8 h-elems)
    32 * WHP +     // sWhh
    33 * VHP +     // sVH   : this block's attention positions
    X2L * HP +     // sHcol : H columns for these positions
    256 +          // sw3
    256 +          // sx2
    256 +          // sh
    512 +          // sx
    256 +          // sb2f  : full b2
    33 +           // sv
    33 +           // sbeta (unnormalized exp)
    32 +           // sgates
    32 +           // sbias (b_ih+b_hh)
    32 +           // sscale: per-block softmax rescale
    8 +            // sc    : private cell state
    8 +            // shn   : new h elements
    4;             // sred
constexpr int SMEM_BYTES = SMEM_FLOATS * 4;

typedef float v2f __attribute__((ext_vector_type(2)));
typedef float v8f __attribute__((ext_vector_type(8)));

// ============================================================
// Kernel A: VH[j,i] = sum_k H[k,j] * W1[i,k]   (fp32 WMMA GEMM)
// ============================================================
__global__ void __launch_bounds__(256) vh_gemm(const float* __restrict__ H,
                                               const float* __restrict__ W1,
                                               float* __restrict__ VH) {
  __shared__ float ldsW[512 * 16];  // W1 panel, transposed [k][n]
  const int t  = threadIdx.x;
  const int n0 = blockIdx.y * 16;
  for (int idx = t; idx < 16 * 512; idx += 256) {
    int n = idx >> 9, k = idx & 511;
    ldsW[k * 16 + n] = W1[(n0 + n) * 512 + k];
  }
  __syncthreads();

  const int lane = t & 31;
  const int w    = t >> 5;
  const int m    = lane & 15;
  const int ks   = (lane >> 4) << 1;          // K offset 0 / 2 per half-wave
  const int jt   = blockIdx.x * 8 + w;
  const int jrow = jt * 16 + m;

  v8f acc = {};
#pragma unroll 4
  for (int k0 = 0; k0 < 512; k0 += 4) {
    v2f a, bb;
    a.x  = H[(k0 + ks) * SEQ + jrow];
    a.y  = H[(k0 + ks + 1) * SEQ + jrow];
    bb.x = ldsW[(k0 + ks) * 16 + m];
    bb.y = ldsW[(k0 + ks + 1) * 16 + m];
    acc = __builtin_amdgcn_wmma_f32_16x16x4_f32(false, a, false, bb,
                                                (short)0, acc, false, false);
  }
  const int mh = (lane >> 4) << 3;
#pragma unroll
  for (int r = 0; r < 8; ++r)
    VH[(jt * 16 + r + mh) * HID + n0 + m] = acc[r];

  if (blockIdx.x == 0 && t < 16) VH[SEQ * HID + n0 + t] = 0.0f;  // terminal row
}

// ============================================================
// CDNA5 async global->LDS copy (ASYNCcnt-tracked, 16B per lane)
// ============================================================
__device__ __forceinline__ void async_g2l_b128(unsigned lds_off, int goff,
                                               const void* base) {
  asm volatile("global_load_async_to_lds_b128 %0, %1, %2"
               :: "v"(lds_off), "v"(goff), "s"(base) : "memory");
}
__device__ __forceinline__ void async_wait0() {
  asm volatile("s_wait_asynccnt 0x0" ::: "memory");
}

// ============================================================
// Grid-wide barrier: monotone arrival counter
// ============================================================
__device__ __forceinline__ void gsync(unsigned* bar, unsigned* cnt) {
  __threadfence();
  __syncthreads();
  if (threadIdx.x == 0) {
    unsigned target = (++(*cnt)) * (unsigned)NBLK;
    __hip_atomic_fetch_add(bar, 1u, __ATOMIC_ACQ_REL, __HIP_MEMORY_SCOPE_AGENT);
    while (__hip_atomic_load(bar, __ATOMIC_ACQUIRE, __HIP_MEMORY_SCOPE_AGENT) < target)
      __builtin_amdgcn_s_sleep(1);
  }
  __syncthreads();
}

// ============================================================
// Kernel B: cooperative scan, 2 grid barriers per step.
// Block b owns positions [32b,32b+32) (+terminal for b=31),
// h-elements [8b,8b+8) with all 4 gate rows each, and W2 columns [8b,8b+8).
// ============================================================
__global__ void __launch_bounds__(256) ptr_scan(
    const float* __restrict__ H,   const float* __restrict__ W2,
    const float* __restrict__ b2,  const float* __restrict__ w3,
    const float* __restrict__ cof, const float* __restrict__ Wih,
    const float* __restrict__ Whh, const float* __restrict__ bih,
    const float* __restrict__ bhh, const float* __restrict__ VH,
    float* __restrict__ g_xpart, float* __restrict__ g_x2p,
    float* __restrict__ g_m, float* __restrict__ g_s,
    float* __restrict__ g_h, unsigned* __restrict__ bar,
    float* __restrict__ out) {
  extern __shared__ float sm[];
  float* sW2T  = sm;
  float* sWih  = sW2T + 8 * W2P;
  float* sWhh  = sWih + 32 * WIP;
  float* sVH   = sWhh + 32 * WHP;
  float* sHcol = sVH + 33 * VHP;
  float* sw3   = sHcol + X2L * HP;
  float* sx2   = sw3 + 256;
  float* sh    = sx2 + 256;
  float* sx    = sh + 256;
  float* sb2f  = sx + 512;
  float* sv    = sb2f + 256;
  float* sbeta = sv + 33;
  float* sgates= sbeta + 33;
  float* sbias = sgates + 32;
  float* sscale= sbias + 32;
  float* sc    = sscale + 32;
  float* shn   = sc + 8;
  float* sred  = shn + 8;

  const int t    = threadIdx.x;
  const int b    = blockIdx.x;
  const int jbeg = b * 32;
  const int jcnt = (b == NBLK - 1) ? 33 : 32;
  const float c0 = cof[0];

  // ---- async-stage the big LDS-resident blocks (vec4 per lane) ----
  {
    const unsigned oWih = (unsigned)(size_t)(void*)sWih;
    for (int idx = t; idx < 32 * 128; idx += T) {
      int rid = idx >> 7, kv = (idx & 127) << 2;
      int grow = (rid >> 3) * 256 + b * 8 + (rid & 7);
      async_g2l_b128(oWih + (unsigned)(rid * WIP + kv) * 4u,
                     (grow * 512 + kv) * 4, Wih);
    }
    const unsigned oWhh = (unsigned)(size_t)(void*)sWhh;
    for (int idx = t; idx < 32 * 64; idx += T) {
      int rid = idx >> 6, kv = (idx & 63) << 2;
      int grow = (rid >> 3) * 256 + b * 8 + (rid & 7);
      async_g2l_b128(oWhh + (unsigned)(rid * WHP + kv) * 4u,
                     (grow * 256 + kv) * 4, Whh);
    }
    const unsigned oVH = (unsigned)(size_t)(void*)sVH;
    for (int idx = t; idx < jcnt * 64; idx += T) {
      int jo = idx >> 6, kv = (idx & 63) << 2;
      async_g2l_b128(oVH + (unsigned)(jo * VHP + kv) * 4u,
                     ((jbeg + jo) * 256 + kv) * 4, VH);
    }
  }
  // ---- plain staging for transposed / small pieces ----
  for (int idx = t; idx < 8 * 256; idx += T) {        // W2 column slice
    int kl = idx & 7, e = idx >> 3;
    sW2T[kl * W2P + e] = W2[e * 256 + b * 8 + kl];
  }
  for (int idx = t; idx < X2L * jcnt; idx += T) {     // H columns
    int k = idx / jcnt, jo = idx - k * jcnt;
    int j = jbeg + jo;
    sHcol[k * HP + jo] = (j < SEQ) ? H[(size_t)k * SEQ + j] : 0.0f;
  }
  sw3[t]  = w3[t];
  sb2f[t] = b2[t];
  if (t < 32) {
    int grow = (t >> 3) * 256 + b * 8 + (t & 7);
    sbias[t] = bih[grow] + bhh[grow];
  }
  if (t < 8) sc[t] = 0.0f;
  if (t < jcnt) out[jbeg + t] = -logf((float)SEQ1);   // output row 0
  async_wait0();
  __syncthreads();

  unsigned barcnt = 0;
  for (int st = 0; st < SEQ; ++st) {
    // ======== Phase A (h and x2-partials published at previous Bc) ========
    sh[t] = g_h[t];
    {
      float acc = sb2f[t];                            // x2 = sum_b partials + b2
#pragma unroll 4
      for (int bb = 0; bb < NBLK; ++bb) acc += g_x2p[bb * HID + t];
      sx2[t] = acc;
    }
    __syncthreads();

    // attention scores: 8 threads per position
    {
      int pos = t >> 3, sub = t & 7;
      float acc = 0.f;
#pragma unroll 4
      for (int i = 0; i < 32; ++i) {
        int k = sub + 8 * i;
        acc += tanhf(sVH[pos * VHP + k] + sx2[k]) * sw3[k];
      }
      for (int o = 4; o; o >>= 1) acc += __shfl_xor(acc, o, 8);
      if (sub == 0) sv[pos] = acc + c0;
    }
    if (jcnt == 33 && t < 8) {                        // terminal position
      float a2 = 0.f;
      for (int i = 0; i < 32; ++i) {
        int k = t + 8 * i;
        a2 += tanhf(sVH[32 * VHP + k] + sx2[k]) * sw3[k];
      }
      for (int o = 4; o; o >>= 1) a2 += __shfl_xor(a2, o, 8);
      if (t == 0) sv[32] = a2 + c0;
    }
    __syncthreads();

    if (t == 0) {                                     // block softmax stats
      float m = -3.4e38f, s = 0.f;
      for (int jo = 0; jo < jcnt; ++jo) m = fmaxf(m, sv[jo]);
      for (int jo = 0; jo < jcnt; ++jo) s += expf(sv[jo] - m);
      g_m[b] = m; g_s[b] = s; sred[1] = m;
    }
    __syncthreads();
    if (t < jcnt) sbeta[t] = expf(sv[t] - sred[1]);   // unnormalized
    __syncthreads();

    for (int k = t; k < X2L; k += T) {                // unnormalized x partial
      float acc = 0.f;
      for (int jo = 0; jo < jcnt; ++jo) acc += sHcol[k * HP + jo] * sbeta[jo];
      g_xpart[b * X2L + k] = acc;
    }
    gsync(bar, &barcnt);   // B1: stats + x partials visible

    // ======== Phase B ========
    if (t == 0) {
      float m = -3.4e38f;
      for (int i = 0; i < NBLK; ++i) m = fmaxf(m, g_m[i]);
      float s = 0.f;
      for (int i = 0; i < NBLK; ++i) s += g_s[i] * expf(g_m[i] - m);
      sred[0] = m + logf(s);                          // logZ
    }
    __syncthreads();
    const float logZ = sred[0];
    if (t < NBLK) sscale[t] = expf(g_m[t] - logZ);
    if (t < jcnt) out[(size_t)(st + 1) * SEQ1 + jbeg + t] = sv[t] - logZ;
    __syncthreads();

    for (int k = t; k < X2L; k += T) {                // gather x
      float acc = 0.f;
#pragma unroll 4
      for (int bb = 0; bb < NBLK; ++bb) acc += g_xpart[bb * X2L + k] * sscale[bb];
      sx[k] = acc;
    }
    __syncthreads();

    // gates: 8 threads per gate row (32 rows)
    {
      int rid = t >> 3, sub = t & 7;
      float acc = 0.f;
#pragma unroll 8
      for (int i = 0; i < 64; ++i) { int k = sub + 8 * i; acc += sWih[rid * WIP + k] * sx[k]; }
#pragma unroll 8
      for (int i = 0; i < 32; ++i) { int k = sub + 8 * i; acc += sWhh[rid * WHP + k] * sh[k]; }
      for (int o = 4; o; o >>= 1) acc += __shfl_xor(acc, o, 8);
      if (sub == 0) sgates[rid] = acc + sbias[rid];
    }
    __syncthreads();

    if (t < 8) {                                      // block-local LSTM update
      float gi = sgates[t], gf = sgates[8 + t], gg = sgates[16 + t], go = sgates[24 + t];
      float si = 1.f / (1.f + expf(-gi));
      float sf = 1.f / (1.f + expf(-gf));
      float so = 1.f / (1.f + expf(-go));
      float cn = sf * sc[t] + si * tanhf(gg);
      float hn = so * tanhf(cn);
      sc[t] = cn; shn[t] = hn;
      g_h[b * 8 + t] = hn;
    }
    __syncthreads();

    {                                                 // x2 partial for next step
      float acc = 0.f;
#pragma unroll
      for (int kl = 0; kl < 8; ++kl) acc += sW2T[kl * W2P + t] * shn[kl];
      g_x2p[b * HID + t] = acc;
    }
    gsync(bar, &barcnt);   // Bc: h + x2 partials visible
  }
}

// ============================================================
extern "C" void kernel_launch(void* const* d_in, const int* in_sizes, int n_in,
                              void* d_out, int out_size, void* d_ws, size_t ws_size,
                              hipStream_t stream) {
  (void)in_sizes; (void)n_in; (void)out_size; (void)ws_size;
  const float* H   = (const float*)d_in[0];
  const float* W1  = (const float*)d_in[1];
  const float* W2  = (const float*)d_in[2];
  const float* b2  = (const float*)d_in[3];
  const float* w3  = (const float*)d_in[4];
  const float* cof = (const float*)d_in[5];
  const float* Wih = (const float*)d_in[6];
  const float* Whh = (const float*)d_in[7];
  const float* bih = (const float*)d_in[8];
  const float* bhh = (const float*)d_in[9];
  float* out = (float*)d_out;

  float* ws      = (float*)d_ws;
  float* VH      = ws;                        // 1025*256
  float* g_xpart = VH + SEQ1 * HID;           // 32*512
  float* g_x2p   = g_xpart + NBLK * X2L;      // 32*256  (memset)
  float* g_m     = g_x2p + NBLK * HID;        // 32
  float* g_s     = g_m + NBLK;                // 32
  float* g_h     = g_s + NBLK;                // 256     (memset)
  unsigned* bar  = (unsigned*)(g_h + HID);    // counter (memset)

  // zero x2-partials + h + barrier counter (graph-capturable)
  hipMemsetAsync(g_x2p, 0,
                 (NBLK * HID + NBLK + NBLK + HID) * sizeof(float) + 64, stream);

  vh_gemm<<<dim3(8, 16, 1), 256, 0, stream>>>(H, W1, VH);

  hipFuncSetAttribute(reinterpret_cast<const void*>(ptr_scan),
                      hipFuncAttributeMaxDynamicSharedMemorySize, SMEM_BYTES);
  ptr_scan<<<dim3(NBLK, 1, 1), T, SMEM_BYTES, stream>>>(
      H, W2, b2, w3, cof, Wih, Whh, bih, bhh, VH,
      g_xpart, g_x2p, g_m, g_s, g_h, bar, out);
}